// Adder12Bit_43860206027277
// MI455X (gfx1250) — compile-verified
//
#include <hip/hip_runtime.h>
#include <cstdint>

// 12-bit ripple-carry adder over {0,1} f32 "bits" — HBM-bound streaming kernel.
// Stage 256-row tiles through LDS via CDNA5 async global<->LDS DMA (ASYNCcnt),
// GVS addressing (SGPR64 base + VGPR32 offset + inst offset applied to both the
// global and LDS sides), compute the add as integer bit-packing, stream results
// back with async LDS->global b128 stores.

#define NBITS 12
#define ROWS 256                        // rows per block == threads per block
#define TILE_FLOATS (ROWS * NBITS)      // 3072 floats
#define TILE_BYTES (TILE_FLOATS * 4)    // 12288 bytes = 3 x 4096

__device__ __forceinline__ unsigned pack4(unsigned acc, float4 v) {
    // inputs are exactly 0.0f (0x00000000) or 1.0f (0x3F800000): bit29 is the flag
    acc = (acc << 1) | ((__float_as_uint(v.x) >> 29) & 1u);
    acc = (acc << 1) | ((__float_as_uint(v.y) >> 29) & 1u);
    acc = (acc << 1) | ((__float_as_uint(v.z) >> 29) & 1u);
    acc = (acc << 1) | ((__float_as_uint(v.w) >> 29) & 1u);
    return acc;
}

__global__ __launch_bounds__(256)
void Adder12Bit_kernel(const float* __restrict__ A,
                       const float* __restrict__ B,
                       float* __restrict__ out,
                       long long batch) {
    // LDS: [A tile | B tile | Sum tile], each an exact contiguous 256x12 f32 tile.
    __shared__ __align__(16) float lds[TILE_FLOATS * 3];

    const int tid = threadIdx.x;
    const long long rowBase = (long long)blockIdx.x * ROWS;
    const unsigned long long tileByte = (unsigned long long)rowBase * (NBITS * 4);

    // Low 32 bits of a flat shared-aperture address == LDS byte offset.
    const unsigned ldsBase = (unsigned)(uintptr_t)(void*)lds;
    const unsigned vOff  = (unsigned)tid * 16u;          // per-lane byte offset
    const unsigned vLdsA = ldsBase + vOff;
    const unsigned vLdsB = ldsBase + TILE_BYTES + vOff;
    const unsigned vLdsS = ldsBase + 2u * TILE_BYTES + vOff;

    // Uniform per-block tile bases -> SGPR pairs (GVS mode saddr).
    const unsigned long long sA = (unsigned long long)(uintptr_t)A + tileByte;
    const unsigned long long sB = (unsigned long long)(uintptr_t)B + tileByte;
    const unsigned long long sO = (unsigned long long)(uintptr_t)out + tileByte;

    // ---- Async stage-in: coalesced 16B chunks, global -> LDS (GVS mode).
    // offset: immediate advances BOTH the global and LDS addresses.
    asm volatile(
        "global_load_async_to_lds_b128 %0, %2, %3\n\t"
        "global_load_async_to_lds_b128 %0, %2, %3 offset:4096\n\t"
        "global_load_async_to_lds_b128 %0, %2, %3 offset:8192\n\t"
        "global_load_async_to_lds_b128 %1, %2, %4\n\t"
        "global_load_async_to_lds_b128 %1, %2, %4 offset:4096\n\t"
        "global_load_async_to_lds_b128 %1, %2, %4 offset:8192\n\t"
        "s_wait_asynccnt 0"
        :: "v"(vLdsA), "v"(vLdsB), "v"(vOff), "s"(sA), "s"(sB)
        : "memory");
    __syncthreads();                       // all waves' tiles visible

    // ---- Compute: pack 12 bits (bit 11 = LSB), add, unpack 13 bits.
    // Row = 48 B = 3 aligned float4s -> ds_load_b128 x3 per matrix.
    const float4* a4 = reinterpret_cast<const float4*>(lds) + tid * 3;
    const float4* b4 = a4 + (TILE_FLOATS / 4);
    unsigned ia = 0u, ib = 0u;
    ia = pack4(ia, a4[0]); ia = pack4(ia, a4[1]); ia = pack4(ia, a4[2]);
    ib = pack4(ib, b4[0]); ib = pack4(ib, b4[1]); ib = pack4(ib, b4[2]);
    const unsigned s = ia + ib;            // 13-bit binary sum

    float4* s4 = reinterpret_cast<float4*>(lds) + 2 * (TILE_FLOATS / 4) + tid * 3;
#pragma unroll
    for (int k = 0; k < 3; ++k) {          // ds_store_b128 x3
        float4 v;
        v.x = (float)((s >> (11 - 4 * k)) & 1u);
        v.y = (float)((s >> (10 - 4 * k)) & 1u);
        v.z = (float)((s >> (9  - 4 * k)) & 1u);
        v.w = (float)((s >> (8  - 4 * k)) & 1u);
        s4[k] = v;
    }
    const float carry = (float)((s >> NBITS) & 1u);

    __syncthreads();                       // sum tile complete across waves

    // ---- Async stage-out: coalesced 16B chunks, LDS -> global (GVS mode).
    asm volatile(
        "global_store_async_from_lds_b128 %1, %0, %2\n\t"
        "global_store_async_from_lds_b128 %1, %0, %2 offset:4096\n\t"
        "global_store_async_from_lds_b128 %1, %0, %2 offset:8192"
        :: "v"(vLdsS), "v"(vOff), "s"(sO)
        : "memory");

    // Carry-out: one float per thread, naturally coalesced.
    out[batch * NBITS + rowBase + tid] = carry;
    // s_endpgm's implicit wait-idle drains the outstanding async stores.
}

extern "C" void kernel_launch(void* const* d_in, const int* in_sizes, int n_in,
                              void* d_out, int out_size, void* d_ws, size_t ws_size,
                              hipStream_t stream) {
    const float* A = (const float*)d_in[0];
    const float* B = (const float*)d_in[1];
    float* out = (float*)d_out;

    const long long batch = (long long)in_sizes[0] / NBITS;  // 4194304
    const int blocks = (int)(batch / ROWS);                  // 16384 (exact)

    Adder12Bit_kernel<<<blocks, ROWS, 0, stream>>>(A, B, out, batch);
}